// ConvEmbeddingXY_850403525144
// MI455X (gfx1250) — compile-verified
//
#include <hip/hip_runtime.h>

#define BATCH 32
#define NPTS  1000
#define KNN   10
#define HDIM  128
#define KPAD  44            // 20 (x-sorted) + 20 (y-sorted) + 2 (px,py) + 2 pad
#define NODES (BATCH * NPTS) // 32000 = 2000 strips of 16, exact

typedef __attribute__((ext_vector_type(2))) float v2f;
typedef __attribute__((ext_vector_type(8))) float v8f;

// ---------------------------------------------------------------------------
// Kernel A: fold conv weights through W2.
//   M[j][g]   = sum_i Wsel[i][c][k] * W2[i][g]   (j = sel*20 + c*10 + k, j<40)
//   M[40..41] = W1 rows;  M[42..43] = 0
//   beta[g]   = sum_i (bx[i]+by[i]) * W2[i][g] + b1[g] + b2[g]
// ---------------------------------------------------------------------------
__global__ void prep_kernel(const float* __restrict__ Wx, const float* __restrict__ bx,
                            const float* __restrict__ Wy, const float* __restrict__ by,
                            const float* __restrict__ W1, const float* __restrict__ b1,
                            const float* __restrict__ W2, const float* __restrict__ b2,
                            float* __restrict__ M, float* __restrict__ beta) {
    int g = threadIdx.x;  // 0..127
    for (int j = 0; j < 40; ++j) {
        const float* W = (j < 20) ? Wx : Wy;
        int jj = j % 20;  // c*10 + k, matches Wsel[i*20 + jj]
        float acc = 0.f;
        for (int i = 0; i < HDIM; ++i)
            acc += W[i * 20 + jj] * W2[i * HDIM + g];
        M[j * HDIM + g] = acc;
    }
    M[40 * HDIM + g] = W1[0 * HDIM + g];
    M[41 * HDIM + g] = W1[1 * HDIM + g];
    M[42 * HDIM + g] = 0.f;
    M[43 * HDIM + g] = 0.f;
    float acc = 0.f;
    for (int i = 0; i < HDIM; ++i)
        acc += (bx[i] + by[i]) * W2[i * HDIM + g];
    beta[g] = acc + b1[g] + b2[g];
}

// ---------------------------------------------------------------------------
// Kernel B: per-node KNN (top-10 incl. self), coordinate sorts, emit G row.
// One batch's coords live in LDS (8 KB). 4 blocks of 256 threads per batch.
// ---------------------------------------------------------------------------
__global__ void knn_feat_kernel(const float* __restrict__ x, float* __restrict__ G) {
    __shared__ float Xs[NPTS];
    __shared__ float Ys[NPTS];
    int bat   = blockIdx.x >> 2;
    int local = (blockIdx.x & 3) * 256 + threadIdx.x;
    const float* xb = x + (size_t)bat * NPTS * 2;
    for (int i = threadIdx.x; i < NPTS; i += 256) {
        Xs[i] = xb[2 * i + 0];
        Ys[i] = xb[2 * i + 1];
    }
    __syncthreads();
    if (local >= NPTS) return;

    float px = Xs[local], py = Ys[local];

    // register-resident sorted top-10 (ascending d2); strict '<' keeps the
    // earliest index on ties, matching top_k's preference for lower indices.
    float nd[KNN]; int ni[KNN];
#pragma unroll
    for (int t = 0; t < KNN; ++t) { nd[t] = 3.0e38f; ni[t] = 0; }
    for (int j = 0; j < NPTS; ++j) {
        float dx = px - Xs[j], dy = py - Ys[j];
        float d  = dx * dx + dy * dy;
        if (d < nd[KNN - 1]) {
            int id = j;
#pragma unroll
            for (int t = 0; t < KNN; ++t) {
                bool  c  = d < nd[t];
                float td = c ? nd[t] : d;
                int   ti = c ? ni[t] : id;
                nd[t] = c ? d  : nd[t];
                ni[t] = c ? id : ni[t];
                d = td; id = ti;
            }
        }
    }

    // gather neighbor coords
    float sx0[KNN], sx1[KNN], sy0[KNN], sy1[KNN];
#pragma unroll
    for (int t = 0; t < KNN; ++t) {
        float cx = Xs[ni[t]], cy = Ys[ni[t]];
        sx0[t] = cx; sx1[t] = cy;   // to be sorted by x
        sy0[t] = cx; sy1[t] = cy;   // to be sorted by y
    }
    // stable bubble sort by x-coordinate
#pragma unroll
    for (int i = 0; i < KNN - 1; ++i)
#pragma unroll
        for (int j = 0; j < KNN - 1 - i; ++j)
            if (sx0[j] > sx0[j + 1]) {
                float t0 = sx0[j]; sx0[j] = sx0[j + 1]; sx0[j + 1] = t0;
                float t1 = sx1[j]; sx1[j] = sx1[j + 1]; sx1[j + 1] = t1;
            }
    // stable bubble sort by y-coordinate
#pragma unroll
    for (int i = 0; i < KNN - 1; ++i)
#pragma unroll
        for (int j = 0; j < KNN - 1 - i; ++j)
            if (sy1[j] > sy1[j + 1]) {
                float t0 = sy0[j]; sy0[j] = sy0[j + 1]; sy0[j + 1] = t0;
                float t1 = sy1[j]; sy1[j] = sy1[j + 1]; sy1[j + 1] = t1;
            }

    // G row layout: [c0 x-sorted k0..9 | c1 x-sorted | c0 y-sorted | c1 y-sorted | px py 0 0]
    float* gr = G + (size_t)(bat * NPTS + local) * KPAD;
#pragma unroll
    for (int k = 0; k < KNN; ++k) {
        gr[0 * KNN + k]      = sx0[k];
        gr[1 * KNN + k]      = sx1[k];
        gr[20 + 0 * KNN + k] = sy0[k];
        gr[20 + 1 * KNN + k] = sy1[k];
    }
    gr[40] = px; gr[41] = py; gr[42] = 0.f; gr[43] = 0.f;
}

// ---------------------------------------------------------------------------
// Kernel C: out[32000 x 128] = G[32000 x 44] @ M[44 x 128] + beta
// fp32 WMMA 16x16x4. One wave = 16-node strip x 128 cols: 8 accum tiles,
// 11 k-steps -> 88 v_wmma_f32_16x16x4_f32 per wave. M/beta staged in LDS.
// ---------------------------------------------------------------------------
__global__ void gemm_wmma_kernel(const float* __restrict__ G,
                                 const float* __restrict__ M,
                                 const float* __restrict__ beta,
                                 float* __restrict__ out) {
    __shared__ float Ms[KPAD * HDIM];  // 22528 B
    __shared__ float Bs[HDIM];
    for (int i = threadIdx.x; i < KPAD * HDIM; i += blockDim.x) Ms[i] = M[i];
    for (int i = threadIdx.x; i < HDIM; i += blockDim.x) Bs[i] = beta[i];
    __syncthreads();

    int wave  = threadIdx.x >> 5;
    int lane  = threadIdx.x & 31;
    int strip = blockIdx.x * 4 + wave;  // 16-node strip
    int base  = strip * 16;
    int half  = lane >> 4;              // 0: K pair {0,1}, 1: K pair {2,3}
    int l15   = lane & 15;
    int koffb = half * 2;

    const v8f vz = {0.f, 0.f, 0.f, 0.f, 0.f, 0.f, 0.f, 0.f};
    v8f acc[8];
#pragma unroll
    for (int t = 0; t < 8; ++t) acc[t] = vz;

    const float* Arow = G + (size_t)(base + l15) * KPAD;
#pragma unroll
    for (int kb = 0; kb < 11; ++kb) {
        int k0 = kb * 4 + koffb;
        v2f a;  // A 16x4: lanes 0-15 -> M=l15, K={k0,k0+1}; lanes 16-31 -> K={k0,k0+1}+2
        a.x = Arow[k0];
        a.y = Arow[k0 + 1];
#pragma unroll
        for (int t = 0; t < 8; ++t) {
            v2f b;  // B 4x16: row-striped across lanes, same K-pair split as A
            b.x = Ms[(k0)     * HDIM + t * 16 + l15];
            b.y = Ms[(k0 + 1) * HDIM + t * 16 + l15];
            acc[t] = __builtin_amdgcn_wmma_f32_16x16x4_f32(
                false, a, false, b, (short)0, acc[t], false, false);
        }
    }

    // D 16x16 f32: VGPR r -> row r (lanes 0-15) / r+8 (lanes 16-31), col = l15
#pragma unroll
    for (int t = 0; t < 8; ++t) {
        int   h  = t * 16 + l15;
        float bv = Bs[h];
#pragma unroll
        for (int r = 0; r < 8; ++r) {
            int node = base + r + half * 8;
            out[(size_t)node * HDIM + h] = acc[t][r] + bv;
        }
    }
}

// ---------------------------------------------------------------------------
extern "C" void kernel_launch(void* const* d_in, const int* in_sizes, int n_in,
                              void* d_out, int out_size, void* d_ws, size_t ws_size,
                              hipStream_t stream) {
    const float* x  = (const float*)d_in[0];
    const float* Wx = (const float*)d_in[1];
    const float* bx = (const float*)d_in[2];
    const float* Wy = (const float*)d_in[3];
    const float* by = (const float*)d_in[4];
    const float* W1 = (const float*)d_in[5];
    const float* b1 = (const float*)d_in[6];
    const float* W2 = (const float*)d_in[7];
    const float* b2 = (const float*)d_in[8];
    float* out = (float*)d_out;

    char*  ws   = (char*)d_ws;
    float* M    = (float*)ws;                             // 44*128 floats
    float* beta = (float*)(ws + KPAD * HDIM * 4);         // 128 floats
    float* G    = (float*)(ws + KPAD * HDIM * 4 + 512);   // 32000*44 floats

    prep_kernel<<<1, HDIM, 0, stream>>>(Wx, bx, Wy, by, W1, b1, W2, b2, M, beta);
    knn_feat_kernel<<<BATCH * 4, 256, 0, stream>>>(x, G);
    gemm_wmma_kernel<<<NODES / 64, 128, 0, stream>>>(G, M, beta, out);
}